// SoftProjection_25744033972896
// MI455X (gfx1250) — compile-verified
//
#include <hip/hip_runtime.h>

typedef __attribute__((ext_vector_type(2))) float v2f;
typedef __attribute__((ext_vector_type(8))) float v8f;

#define B_   8
#define N_   8192
#define M_   2048
#define KNN  16
#define WAVES_PER_BLOCK 8

// Branchless shift-insert into ascending sorted list; no-op when v >= s[KNN-1].
__device__ __forceinline__ void topk_insert(float (&s)[KNN], int (&si)[KNN],
                                            float v, int vi) {
#pragma unroll
    for (int i = 0; i < KNN; ++i) {
        const bool  lt = v < s[i];
        const float ts = s[i];
        const int   ti = si[i];
        s[i]  = lt ? v  : ts;
        si[i] = lt ? vi : ti;
        v  = lt ? ts : v;
        vi = lt ? ti : vi;
    }
}

// Process one WMMA result tile: min-tree filter first, per-candidate only on hit.
__device__ __forceinline__ void topk_scan(float (&s)[KNN], int (&si)[KNN],
                                          const v8f& d, int base) {
    // 7-op min tree over the 8 accumulator values in this lane
    const float m01 = fminf(d[0], d[1]);
    const float m23 = fminf(d[2], d[3]);
    const float m45 = fminf(d[4], d[5]);
    const float m67 = fminf(d[6], d[7]);
    const float m03 = fminf(m01, m23);
    const float m47 = fminf(m45, m67);
    const float m8  = fminf(m03, m47);
    if (__any(m8 < s[KNN - 1])) {           // rare past warm-up
#pragma unroll
        for (int r = 0; r < 8; ++r) {
            const float sc = d[r];
            if (__any(sc < s[KNN - 1]))
                topk_insert(s, si, sc, base + r);
        }
    }
}

// One wave handles one (batch, 16-query tile).
// Distance scores via V_WMMA_F32_16X16X4_F32:
//   A (16x4, points):  row m = (px, py, pz, |p|^2)
//   B (4x16, queries): col n = (-2qx, -2qy, -2qz, 1)^T
//   D[m][n] = |p_m|^2 - 2 p_m . q_n   (|q|^2 omitted: cancels in top-k & softmax)
// C/D layout: lane L<16 holds query L, point rows 0..7 in acc[0..7];
//             lane L+16 holds query L, point rows 8..15.
__global__ __launch_bounds__(32 * WAVES_PER_BLOCK)
void softproj_wmma_kernel(const float* __restrict__ pc,   // (B,3,N)
                          const float* __restrict__ qc,   // (B,3,M)
                          const float* __restrict__ temp, // scalar
                          float* __restrict__ out) {      // (B,3,M)
    __shared__ float lds_s[WAVES_PER_BLOCK * 32 * KNN];
    __shared__ int   lds_i[WAVES_PER_BLOCK * 32 * KNN];

    const int lane = threadIdx.x & 31;
    const int wave = threadIdx.x >> 5;
    const int tile = blockIdx.x * WAVES_PER_BLOCK + wave;  // 0 .. B*(M/16)-1
    const int b    = tile >> 7;                            // tile / (M/16)
    const int qt   = tile & 127;
    const int qbase = qt * 16;

    const float* pcb = pc + (size_t)b * 3 * N_;
    const float* qcb = qc + (size_t)b * 3 * M_;

    const int half = lane >> 4;   // 0: K=0,1 (A) / rows 0..7 (D); 1: K=2,3 / rows 8..15
    const int l15  = lane & 15;
    const int rowoff = half * 8;

    // ---- B operand (queries), loop-invariant ----
    // VGPR0: lanes 0-15 -> K=0 (-2qx), lanes 16-31 -> K=2 (-2qz)
    // VGPR1: lanes 0-15 -> K=1 (-2qy), lanes 16-31 -> K=3 (1.0)
    const int qi = qbase + l15;
    const float qx = qcb[qi];
    const float qy = qcb[M_ + qi];
    const float qz = qcb[2 * M_ + qi];
    v2f bm;
    bm[0] = half ? (-2.0f * qz) : (-2.0f * qx);
    bm[1] = half ? 1.0f         : (-2.0f * qy);

    // ---- per-lane top-16 (sorted ascending, s[15] = current threshold) ----
    float s[KNN];
    int   si[KNN];
#pragma unroll
    for (int i = 0; i < KNN; ++i) { s[i] = 3.0e38f; si[i] = 0; }

    // ---- stream point tiles, 2 WMMAs (32 points) per iteration ----
    for (int nb = 0; nb < N_; nb += 32) {
        const int pi0 = nb + l15;
        const int pi1 = pi0 + 16;
        const float px0 = pcb[pi0];
        const float py0 = pcb[N_ + pi0];
        const float pz0 = pcb[2 * N_ + pi0];
        const float px1 = pcb[pi1];
        const float py1 = pcb[N_ + pi1];
        const float pz1 = pcb[2 * N_ + pi1];

        // A layout: lanes 0-15: VGPR0=K0(px), VGPR1=K1(py)
        //           lanes16-31: VGPR0=K2(pz), VGPR1=K3(|p|^2)
        v2f a0, a1;
        a0[0] = half ? pz0 : px0;
        a0[1] = half ? (px0 * px0 + py0 * py0 + pz0 * pz0) : py0;
        a1[0] = half ? pz1 : px1;
        a1[1] = half ? (px1 * px1 + py1 * py1 + pz1 * pz1) : py1;

        v8f c0 = {};
        v8f c1 = {};
        const v8f d0 = __builtin_amdgcn_wmma_f32_16x16x4_f32(
            false, a0, false, bm, (short)0, c0, false, false);
        const v8f d1 = __builtin_amdgcn_wmma_f32_16x16x4_f32(
            false, a1, false, bm, (short)0, c1, false, false);

        topk_scan(s, si, d0, nb + rowoff);
        topk_scan(s, si, d1, nb + 16 + rowoff);
    }

    // ---- merge lane L (rows 0..7 stream) with lane L+16 (rows 8..15 stream) ----
    {
        float* ms = &lds_s[(wave * 32 + lane) * KNN];
        int*   mi = &lds_i[(wave * 32 + lane) * KNN];
#pragma unroll
        for (int i = 0; i < KNN; ++i) { ms[i] = s[i]; mi[i] = si[i]; }
    }
    __syncthreads();

    if (lane < 16) {
        const float* ps  = &lds_s[(wave * 32 + lane + 16) * KNN];
        const int*   pi2 = &lds_i[(wave * 32 + lane + 16) * KNN];
#pragma unroll
        for (int k = 0; k < KNN; ++k)
            topk_insert(s, si, ps[k], pi2[k]);

        // ---- softmax weights + projection ----
        const float t   = temp[0];
        const float sig = fmaxf(t * t, 1e-4f) + 1e-8f;
        const float inv = 1.0f / sig;
        const float smin = s[0];   // sorted ascending -> max of -dist == -s[0]
        float wsum = 0.0f, ax = 0.0f, ay = 0.0f, az = 0.0f;
#pragma unroll
        for (int k = 0; k < KNN; ++k) {
            const float w  = __expf(-(s[k] - smin) * inv);
            const int   id = si[k];
            ax += w * pcb[id];
            ay += w * pcb[N_ + id];
            az += w * pcb[2 * N_ + id];
            wsum += w;
        }
        const float r = 1.0f / wsum;
        const int q = qbase + lane;
        out[((size_t)b * 3 + 0) * M_ + q] = ax * r;
        out[((size_t)b * 3 + 1) * M_ + q] = ay * r;
        out[((size_t)b * 3 + 2) * M_ + q] = az * r;
    }
}

extern "C" void kernel_launch(void* const* d_in, const int* in_sizes, int n_in,
                              void* d_out, int out_size, void* d_ws, size_t ws_size,
                              hipStream_t stream) {
    (void)in_sizes; (void)n_in; (void)out_size; (void)d_ws; (void)ws_size;
    const float* pc   = (const float*)d_in[0];
    const float* qc   = (const float*)d_in[1];
    const float* temp = (const float*)d_in[2];
    float* out = (float*)d_out;

    const int total_waves = B_ * (M_ / 16);               // 1024
    dim3 grid(total_waves / WAVES_PER_BLOCK);             // 128 blocks
    dim3 block(32 * WAVES_PER_BLOCK);                     // 256 threads = 8 waves
    hipLaunchKernelGGL(softproj_wmma_kernel, grid, block, 0, stream,
                       pc, qc, temp, out);
}